// QLSTM_65481071396688
// MI455X (gfx1250) — compile-verified
//
#include <hip/hip_runtime.h>
#include <hip/hip_bf16.h>

#define T_STEPS 256
#define B_BATCH 512
#define D_IN    256
#define H_Q     4
#define NCOL    16                    // 4 gates * 4 wires
#define M_ROWS  (T_STEPS * B_BATCH)   // 131072 rows
#define M_TILES (M_ROWS / 16)         // 8192 16-row tiles

typedef __attribute__((ext_vector_type(2))) float v2f;
typedef __attribute__((ext_vector_type(8))) float v8f;

// ---------------------------------------------------------------------------
// Phase 1: Xp[t*B+b][g*4+w] = sum_d inputs[t,b,d] * W_g[w,d]   (d < 256)
// M=131072, K=256, N=16 -> one v_wmma_f32_16x16x4_f32 tile per wave,
// K swept in chunks of 4. Weights staged in LDS once per block, pair-
// interleaved so each lane's B fragment is one ds_load_b64.
// ---------------------------------------------------------------------------
__global__ __launch_bounds__(256) void qlstm_xproj_wmma(
    const float* __restrict__ x,                      // [M_ROWS, 256]
    const float* __restrict__ Wf, const float* __restrict__ Wi,
    const float* __restrict__ Wu, const float* __restrict__ Wo,
    float* __restrict__ Xp)                           // [M_ROWS, 16]
{
    // Blds[(k>>1)*32 + n*2 + (k&1)] = W_{n/4}[n%4][k]  (pairs of k contiguous)
    __shared__ float Blds[D_IN * NCOL];

    const int tid = threadIdx.x;
    {
        const float* Ws[4] = {Wf, Wi, Wu, Wo};
        const int k = tid;                            // 256 threads -> k = 0..255
#pragma unroll
        for (int g = 0; g < 4; ++g) {
#pragma unroll
            for (int w = 0; w < 4; ++w) {
                const int n = g * 4 + w;
                Blds[(k >> 1) * 32 + n * 2 + (k & 1)] = Ws[g][w * 260 + k];
            }
        }
    }
    __syncthreads();

    const int wave = tid >> 5;
    const int lane = tid & 31;
    const int tile = blockIdx.x * 8 + wave;           // 16-row M tile
    const int half = lane >> 4;                       // 0 | 1
    const int l16  = lane & 15;

    // A fragment: a[r] = x[tile*16 + l16][k0 + 2*half + r]
    const float* arow = x + (size_t)(tile * 16 + l16) * D_IN + 2 * half;
    // B fragment: b[r] = W^T[k0 + 2*half + r][l16]
    const float* brow = &Blds[half * 32 + l16 * 2];

    v8f acc = {};
#pragma unroll 8
    for (int k0 = 0; k0 < D_IN; k0 += 4) {
        v2f a = *(const v2f*)(arow + k0);
        v2f b = *(const v2f*)(brow + (k0 >> 1) * 32);
        // (neg_a, A, neg_b, B, c_mod, C, reuse_a, reuse_b)
        acc = __builtin_amdgcn_wmma_f32_16x16x4_f32(
            false, a, false, b, (short)0, acc, false, false);
    }

    // D layout: d[r] -> row (r + 8*half), col l16
#pragma unroll
    for (int r = 0; r < 8; ++r)
        Xp[((size_t)tile * 16 + r + 8 * half) * NCOL + l16] = acc[r];
}

// ---------------------------------------------------------------------------
// Phase 2: sequential LSTM scan. Each thread owns one batch element.
// qlayer collapses to: q[idx] = prod_w (bit(idx,3-w) ? sin^2 : cos^2),
// composed CNOT-ring permutation cidx (precomputed), signed sums for <Z_w>.
// ---------------------------------------------------------------------------
__device__ __forceinline__ float sigf(float v) { return 1.0f / (1.0f + __expf(-v)); }

__global__ __launch_bounds__(128) void qlstm_scan(
    const float* __restrict__ Xp,                     // [T*B, 16]
    const float* __restrict__ Wf, const float* __restrict__ Wi,
    const float* __restrict__ Wu, const float* __restrict__ Wo,
    const float* __restrict__ thf, const float* __restrict__ thi,
    const float* __restrict__ thu, const float* __restrict__ tho,
    float* __restrict__ outs,                         // [T, B, 4]
    float* __restrict__ hx, float* __restrict__ cx)   // [B, 4] each
{
    const int b = blockIdx.x * blockDim.x + threadIdx.x;
    if (b >= B_BATCH) return;

    const float* Ws[4]  = {Wf, Wi, Wu, Wo};
    const float* ths[4] = {thf, thi, thu, tho};
    float Wh[4][4][4];                                // [gate][wire][j]: W[:,256+j]
    float th[4][4];
#pragma unroll
    for (int g = 0; g < 4; ++g) {
#pragma unroll
        for (int w = 0; w < 4; ++w) {
            th[g][w] = ths[g][w];
#pragma unroll
            for (int j = 0; j < 4; ++j)
                Wh[g][w][j] = Ws[g][w * 260 + 256 + j];
        }
    }

    float h[4] = {0.f, 0.f, 0.f, 0.f};
    float c[4] = {0.f, 0.f, 0.f, 0.f};
    // composed CNOT ring: p0(p1(p2(p3(j))))
    static constexpr int CIDX[16] = {0,13,3,14,6,11,5,8,12,1,15,2,10,7,9,4};

    for (int t = 0; t < T_STEPS; ++t) {
        const float* xp = Xp + ((size_t)t * B_BATCH + b) * NCOL;
        if (t + 1 < T_STEPS)
            __builtin_prefetch(Xp + ((size_t)(t + 1) * B_BATCH + b) * NCOL, 0, 3);

        float xr[16];
#pragma unroll
        for (int q4 = 0; q4 < 4; ++q4) {
            float4 v = *(const float4*)(xp + q4 * 4);
            xr[q4*4+0] = v.x; xr[q4*4+1] = v.y; xr[q4*4+2] = v.z; xr[q4*4+3] = v.w;
        }

        float E[4][4];                                // [gate][wire] expectation
#pragma unroll
        for (int g = 0; g < 4; ++g) {
            float c2[4], s2[4];
#pragma unroll
            for (int w = 0; w < 4; ++w) {
                float pre = xr[g * 4 + w];
#pragma unroll
                for (int j = 0; j < 4; ++j) pre += Wh[g][w][j] * h[j];
                const float a = 0.5f * (pre + th[g][w]);
                float sn, cn;
                __sincosf(a, &sn, &cn);
                s2[w] = sn * sn; c2[w] = cn * cn;
            }
            // bits (3,2) -> wires 0,1 ; bits (1,0) -> wires 2,3
            const float p01[4] = { c2[0]*c2[1], c2[0]*s2[1], s2[0]*c2[1], s2[0]*s2[1] };
            const float p23[4] = { c2[2]*c2[3], c2[2]*s2[3], s2[2]*c2[3], s2[2]*s2[3] };
            float e0 = 0.f, e1 = 0.f, e2 = 0.f, e3 = 0.f;
#pragma unroll
            for (int j = 0; j < 16; ++j) {
                const int i = CIDX[j];
                const float q = p01[i >> 2] * p23[i & 3];
                e0 += ((j >> 3) & 1) ? -q : q;
                e1 += ((j >> 2) & 1) ? -q : q;
                e2 += ((j >> 1) & 1) ? -q : q;
                e3 += ( j       & 1) ? -q : q;
            }
            E[g][0] = e0; E[g][1] = e1; E[g][2] = e2; E[g][3] = e3;
        }

        float hn[4];
#pragma unroll
        for (int w = 0; w < 4; ++w) {
            const float fv = sigf(E[0][w]);
            const float iv = sigf(E[1][w]);
            const float gv = tanhf(E[2][w]);
            const float ov = sigf(E[3][w]);
            c[w] = fv * c[w] + iv * gv;
            hn[w] = ov * tanhf(c[w]);
            h[w] = hn[w];
        }
        *(float4*)(outs + ((size_t)t * B_BATCH + b) * 4) =
            make_float4(hn[0], hn[1], hn[2], hn[3]);
    }
    *(float4*)(hx + b * 4) = make_float4(h[0], h[1], h[2], h[3]);
    *(float4*)(cx + b * 4) = make_float4(c[0], c[1], c[2], c[3]);
}

// ---------------------------------------------------------------------------
extern "C" void kernel_launch(void* const* d_in, const int* in_sizes, int n_in,
                              void* d_out, int out_size, void* d_ws, size_t ws_size,
                              hipStream_t stream) {
    const float* x   = (const float*)d_in[0];
    const float* Wf  = (const float*)d_in[1];
    const float* Wi  = (const float*)d_in[2];
    const float* Wu  = (const float*)d_in[3];
    const float* Wo  = (const float*)d_in[4];
    const float* thf = (const float*)d_in[5];
    const float* thi = (const float*)d_in[6];
    const float* thu = (const float*)d_in[7];
    const float* tho = (const float*)d_in[8];

    float* Xp = (float*)d_ws;                         // 131072 * 16 * 4B = 8 MB

    float* outs = (float*)d_out;                      // [T, B, 4]
    float* hx   = outs + (size_t)T_STEPS * B_BATCH * H_Q;
    float* cx   = hx + (size_t)B_BATCH * H_Q;

    qlstm_xproj_wmma<<<M_TILES / 8, 256, 0, stream>>>(x, Wf, Wi, Wu, Wo, Xp);
    qlstm_scan<<<B_BATCH / 128, 128, 0, stream>>>(Xp, Wf, Wi, Wu, Wo,
                                                  thf, thi, thu, tho,
                                                  outs, hx, cx);
}